// ZeroGCN_24988119728777
// MI455X (gfx1250) — compile-verified
//
#include <hip/hip_runtime.h>
#include <hip/hip_bf16.h>

typedef float v2f __attribute__((ext_vector_type(2)));
typedef float v8f __attribute__((ext_vector_type(8)));

#define DIM 128

// ---------------------------------------------------------------------------
// Degree / normalization kernels
// ---------------------------------------------------------------------------
__global__ __launch_bounds__(256) void gcn_deg_init(float* __restrict__ deg, int n) {
    int i = blockIdx.x * 256 + threadIdx.x;
    if (i < n) deg[i] = 1.0f;  // self-loop contributes 1 to every node's degree
}

__global__ __launch_bounds__(256) void gcn_deg_scatter(const int* __restrict__ dst,
                                                       float* __restrict__ deg, int ne) {
    int e = blockIdx.x * 256 + threadIdx.x;
    if (e < ne) unsafeAtomicAdd(&deg[dst[e]], 1.0f);
}

__global__ __launch_bounds__(256) void gcn_deg_rsqrt(float* __restrict__ deg, int n) {
    int i = blockIdx.x * 256 + threadIdx.x;
    if (i < n) deg[i] = rsqrtf(deg[i]);  // deg buffer becomes dinv in place
}

// ---------------------------------------------------------------------------
// Dense GEMM via V_WMMA_F32_16X16X4_F32.
// One wave computes one 16x16 output tile; block = 8 waves = one 16-row strip
// across all 128 output columns. K = 128 fully unrolled -> 32 WMMAs per wave.
// ---------------------------------------------------------------------------
template <bool RELU>
__global__ __launch_bounds__(256) void gcn_gemm_wmma(const float* __restrict__ A,  // [n, 128]
                                                     const float* __restrict__ W,  // [128, 128]
                                                     float* __restrict__ Y,        // [n, 128]
                                                     int nrows) {
    const int wave = threadIdx.x >> 5;       // 0..7 -> 16-col tile index
    const int lane = threadIdx.x & 31;
    const int tileM = blockIdx.x;            // 16-row strip
    const int l15 = lane & 15;
    const int khi = (lane >> 4) << 1;        // 0 (lanes 0-15) or 2 (lanes 16-31)

    int m = (tileM << 4) + l15;              // A-fragment row for this lane
    int mload = m < nrows ? m : nrows - 1;   // clamp (keeps EXEC full for WMMA)
    const float* arow = A + (size_t)mload * DIM;
    const int n = (wave << 4) + l15;         // output column for B/C/D fragments

    v8f acc = {};
#pragma unroll
    for (int k = 0; k < DIM; k += 4) {
        // A 16x4 fragment: VGPR0/1 = A[m][k+khi], A[m][k+khi+1]  (contiguous -> float2)
        v2f a = *(const v2f*)(arow + k + khi);
        if (RELU) {
            a.x = fmaxf(a.x, 0.0f);
            a.y = fmaxf(a.y, 0.0f);
        }
        // B 4x16 fragment: rows k+khi, k+khi+1 striped across lanes 0-15 / 16-31
        v2f b;
        b.x = W[(size_t)(k + khi) * DIM + n];
        b.y = W[(size_t)(k + khi + 1) * DIM + n];
        acc = __builtin_amdgcn_wmma_f32_16x16x4_f32(false, a, false, b, (short)0, acc,
                                                    false, false);
    }

    // D layout: VGPR r -> row (r + 8*(lane>=16)), col = lane&15
    const int mrow0 = (tileM << 4) + ((lane >> 4) << 3);
#pragma unroll
    for (int r = 0; r < 8; ++r) {
        int mr = mrow0 + r;
        if (mr < nrows) Y[(size_t)mr * DIM + n] = acc[r];
    }
}

// ---------------------------------------------------------------------------
// Self-loop initialization: AGG[i,:] = dinv[i]^2 * H[i,:]  (plain stores,
// doubles as the zero/base-init of the accumulation buffer)
// ---------------------------------------------------------------------------
__global__ __launch_bounds__(256) void gcn_selfloop_init(const float* __restrict__ dinv,
                                                         const float* __restrict__ H,
                                                         float* __restrict__ AGG, int n) {
    int t = blockIdx.x * 256 + threadIdx.x;          // one thread per float4
    int node = t >> 5;
    if (node >= n) return;
    int c4 = (t & 31) << 2;
    float dv = dinv[node];
    float s = dv * dv;
    const float4 h = *(const float4*)(H + (size_t)node * DIM + c4);
    float4 o;
    o.x = h.x * s; o.y = h.y * s; o.z = h.z * s; o.w = h.w * s;
    *(float4*)(AGG + (size_t)node * DIM + c4) = o;
}

// ---------------------------------------------------------------------------
// Edge aggregation: one wave per edge, lane handles 4 of the 128 features.
// AGG[dst,:] += dinv[src]*dinv[dst] * H[src,:]  via hardware f32 atomics.
// ---------------------------------------------------------------------------
__global__ __launch_bounds__(256) void gcn_edge_agg(const int* __restrict__ src,
                                                    const int* __restrict__ dst,
                                                    const float* __restrict__ dinv,
                                                    const float* __restrict__ H,
                                                    float* __restrict__ AGG, int ne) {
    int e = blockIdx.x * 8 + (threadIdx.x >> 5);
    if (e >= ne) return;
    int lane = threadIdx.x & 31;
    int s = src[e];
    int d = dst[e];
    float nrm = dinv[s] * dinv[d];
    const float4 v = *(const float4*)(H + (size_t)s * DIM + (lane << 2));
    float* out = AGG + (size_t)d * DIM + (lane << 2);
    unsafeAtomicAdd(out + 0, v.x * nrm);
    unsafeAtomicAdd(out + 1, v.y * nrm);
    unsafeAtomicAdd(out + 2, v.z * nrm);
    unsafeAtomicAdd(out + 3, v.w * nrm);
}

// ---------------------------------------------------------------------------
// Launch
// ---------------------------------------------------------------------------
extern "C" void kernel_launch(void* const* d_in, const int* in_sizes, int n_in,
                              void* d_out, int out_size, void* d_ws, size_t ws_size,
                              hipStream_t stream) {
    const float* x = (const float*)d_in[0];          // [N, 128]
    const int* edge_index = (const int*)d_in[1];     // [2, E]
    const float* W1 = (const float*)d_in[2];         // [128, 128]
    const float* W2 = (const float*)d_in[3];         // [128, 128]
    float* out = (float*)d_out;                      // [N, 128]

    const int N = in_sizes[0] / DIM;
    const int E = in_sizes[1] / 2;
    const int* src = edge_index;
    const int* dst = edge_index + E;

    // Workspace layout: dinv (N floats) | H (N*128) | AGG (N*128)
    char* ws = (char*)d_ws;
    size_t degBytes = ((size_t)N * sizeof(float) + 511) & ~(size_t)511;
    float* dinv = (float*)ws;
    float* H = (float*)(ws + degBytes);
    float* AGG = H + (size_t)N * DIM;

    const int tilesM = (N + 15) / 16;
    const dim3 blk(256);

    // Degrees -> dinv
    gcn_deg_init<<<dim3((N + 255) / 256), blk, 0, stream>>>(dinv, N);
    gcn_deg_scatter<<<dim3((E + 255) / 256), blk, 0, stream>>>(dst, dinv, E);
    gcn_deg_rsqrt<<<dim3((N + 255) / 256), blk, 0, stream>>>(dinv, N);

    // Layer 1: H = X @ W1 ; AGG = D^-1/2 (A+I) D^-1/2 H
    gcn_gemm_wmma<false><<<dim3(tilesM), blk, 0, stream>>>(x, W1, H, N);
    gcn_selfloop_init<<<dim3((N * 32 + 255) / 256), blk, 0, stream>>>(dinv, H, AGG, N);
    gcn_edge_agg<<<dim3((E + 7) / 8), blk, 0, stream>>>(src, dst, dinv, H, AGG, E);

    // Layer 2: H = relu(AGG) @ W2 (ReLU fused into GEMM A-loads; reuse H buf)
    gcn_gemm_wmma<true><<<dim3(tilesM), blk, 0, stream>>>(AGG, W2, H, N);
    gcn_selfloop_init<<<dim3((N * 32 + 255) / 256), blk, 0, stream>>>(dinv, H, out, N);
    gcn_edge_agg<<<dim3((E + 7) / 8), blk, 0, stream>>>(src, dst, dinv, H, out, E);
}